// InversePreEmphasis_47614007443589
// MI455X (gfx1250) — compile-verified
//
#include <hip/hip_runtime.h>
#include <hip/hip_bf16.h>

// h_t = tanh(x_t + 0.97*h_{t-1}), B=32 rows, T=262144, fp32.
// Chunked-scan parallelization exploiting the contraction |dh_t/dh_{t-1}| <= 0.97:
// 512-step warm-up bounds chunk-boundary error by 0.97^512 ~= 1.7e-7 worst case
// (realistically ~0.5^512 -> bit-exact merge), letting 65536 chains of 640
// sequential steps replace one chain of 262144.

#define B_ROWS 32
#define T_LEN  262144
#define CHUNK  128
#define WARM   512
#define NCHUNK (T_LEN / CHUNK)   // 2048 chunks per row
#define BLOCK  256
#define COEF   0.97f

__device__ __forceinline__ float htanh(float a) {
#if __has_builtin(__builtin_amdgcn_tanhf)
    return __builtin_amdgcn_tanhf(a);           // gfx1250 v_tanh_f32, compiler handles hazards
#else
    float r;
    asm("v_tanh_f32 %0, %1\n\t"
        "s_delay_alu instid0(TRANS32_DEP_1)"
        : "=v"(r) : "v"(a));
    return r;
#endif
}

__global__ void __launch_bounds__(BLOCK)
inverse_preemphasis_chunked(const float* __restrict__ x, float* __restrict__ y) {
    const int b = blockIdx.y;                              // row 0..31
    const int c = blockIdx.x * BLOCK + threadIdx.x;        // chunk 0..2047
    const float* __restrict__ xr = x + (size_t)b * T_LEN;
    float*       __restrict__ yr = y + (size_t)b * T_LEN;

    const int t0 = c * CHUNK;
    int ws = t0 - WARM;
    if (ws < 0) ws = 0;

    // gfx1250 global_prefetch_b8: warm L2 for this lane's warm-up + output region
    __builtin_prefetch(xr + ws, 0, 1);
    __builtin_prefetch(xr + t0, 0, 1);

    float h = 0.0f;

    // Warm-up: recurrence without stores. ws, t0 are multiples of 128 ->
    // 16B-aligned float4 loads (global_load_b128).
    for (int t = ws; t < t0; t += 4) {
        const float4 v = *(const float4*)(xr + t);
        h = htanh(__builtin_fmaf(COEF, h, v.x));
        h = htanh(__builtin_fmaf(COEF, h, v.y));
        h = htanh(__builtin_fmaf(COEF, h, v.z));
        h = htanh(__builtin_fmaf(COEF, h, v.w));
    }

    // Output region: recurrence + b128 stores.
    for (int t = t0; t < t0 + CHUNK; t += 4) {
        const float4 v = *(const float4*)(xr + t);
        float4 o;
        o.x = h = htanh(__builtin_fmaf(COEF, h, v.x));
        o.y = h = htanh(__builtin_fmaf(COEF, h, v.y));
        o.z = h = htanh(__builtin_fmaf(COEF, h, v.z));
        o.w = h = htanh(__builtin_fmaf(COEF, h, v.w));
        *(float4*)(yr + t) = o;
    }
}

extern "C" void kernel_launch(void* const* d_in, const int* in_sizes, int n_in,
                              void* d_out, int out_size, void* d_ws, size_t ws_size,
                              hipStream_t stream) {
    (void)in_sizes; (void)n_in; (void)out_size; (void)d_ws; (void)ws_size;
    const float* x = (const float*)d_in[0];
    float* y = (float*)d_out;
    inverse_preemphasis_chunked<<<dim3(NCHUNK / BLOCK, B_ROWS), dim3(BLOCK), 0, stream>>>(x, y);
}